// MultiHeadAttention_39238821216743
// MI455X (gfx1250) — compile-verified
//
#include <hip/hip_runtime.h>
#include <hip/hip_bf16.h>

typedef __attribute__((ext_vector_type(2))) float v2f;
typedef __attribute__((ext_vector_type(8))) float v8f;

// D = A(16x4 f32) * B(4x16 f32) + C(16x16 f32), full fp32 accuracy.
__device__ __forceinline__ v8f wmma4(v2f a, v2f b, v8f c) {
  return __builtin_amdgcn_wmma_f32_16x16x4_f32(false, a, false, b, (short)0, c,
                                               false, false);
}

// ---- CDNA5 async global->LDS path (guarded: falls back to sync staging) ----
#if __has_builtin(__builtin_amdgcn_global_load_async_to_lds_b128) && \
    __has_builtin(__builtin_amdgcn_s_wait_asynccnt)
#define USE_ASYNC 1
#else
#define USE_ASYNC 0
#endif

#if USE_ASYNC
// Builtin signature (from hipcc diagnostic): param0 is
//   __attribute__((address_space(1))) int __attribute__((vector_size(16)))*
// LDS aperture: low 32 bits of the flat address are the LDS byte offset
// (ISA 10.2), so integer round-trips to addrspace(3)/(1) preserve values.
typedef int vi4_t __attribute__((vector_size(16)));
typedef __attribute__((address_space(1))) vi4_t* gptr16_t;
typedef __attribute__((address_space(3))) vi4_t* lptr16_t;

__device__ __forceinline__ void async_ld16(void* lds, const void* g) {
  __builtin_amdgcn_global_load_async_to_lds_b128(
      (gptr16_t)(unsigned long long)g,
      (lptr16_t)(unsigned int)(unsigned long long)lds, 0, 0);
}
#define ASYNC_WAIT() __builtin_amdgcn_s_wait_asynccnt(0)
#endif

#define EMBED 1024
#define NHEAD 16
#define HDIM 64
#define BATCH 4
#define SEQ 2048
#define MTOT (BATCH * SEQ)  // 8192

// ---------------------------------------------------------------------------
// GEMM: out[m,n] = sum_k A[m,k] * W[n,k] + bias[n]
// Tile: 128(M) x 64(N) per 256-thread block; 8 waves as 4(M) x 2(N),
// each wave computes a 32x32 patch = 2x2 WMMA 16x16 tiles.
// Async path: double-buffered LDS, one barrier per K-chunk.
// split_heads==1 scatters output to [B, H, L, D] layout.
// ---------------------------------------------------------------------------
__global__ __launch_bounds__(256) void gemm_wmma(
    const float* __restrict__ A,     // [M, K] row-major
    const float* __restrict__ W,     // [N, K] row-major
    const float* __restrict__ bias,  // [N]
    float* __restrict__ out, int M, int N, int K, int split_heads) {
  __shared__ float xs[2][128][20];  // pad 20: conflict-free + 16B aligned
  __shared__ float wsh[2][64][20];

  const int t = threadIdx.x;
  const int lane = t & 31, wid = t >> 5;
  const int l16 = lane & 15, lh = lane >> 4;
  const int wm = wid & 3, wn = wid >> 2;
  const int mBase = blockIdx.y * 128;
  const int nBase = blockIdx.x * 64;

  const int xrow = t >> 1, xcb = (t & 1) * 8;  // A tile 128x16, 8 floats/thr
  const int wrow = t >> 2, wcb = (t & 3) * 4;  // W tile 64x16, 4 floats/thr

  v8f acc[2][2] = {};

  auto compute = [&](int buf) {
#pragma unroll
    for (int kc = 0; kc < 4; ++kc) {
      const int k0 = kc * 4 + lh * 2;  // K index for reg0 (reg1 = k0+1)
      v2f a[2], b[2];
#pragma unroll
      for (int mi = 0; mi < 2; ++mi) {
        int r = wm * 32 + mi * 16 + l16;  // A: M = lane (both halves)
        a[mi][0] = xs[buf][r][k0];
        a[mi][1] = xs[buf][r][k0 + 1];
      }
#pragma unroll
      for (int ni = 0; ni < 2; ++ni) {
        int c = wn * 32 + ni * 16 + l16;  // B: N = lane, value = W[n][k]
        b[ni][0] = wsh[buf][c][k0];
        b[ni][1] = wsh[buf][c][k0 + 1];
      }
#pragma unroll
      for (int mi = 0; mi < 2; ++mi)
#pragma unroll
        for (int ni = 0; ni < 2; ++ni)
          acc[mi][ni] = wmma4(a[mi], b[ni], acc[mi][ni]);
    }
  };

#if USE_ASYNC
  auto stage = [&](int buf, int kk) {
    const float* xsrc = A + (size_t)(mBase + xrow) * K + kk + xcb;
    async_ld16(&xs[buf][xrow][xcb], xsrc);
    async_ld16(&xs[buf][xrow][xcb + 4], xsrc + 4);
    const float* wsrc = W + (size_t)(nBase + wrow) * K + kk + wcb;
    async_ld16(&wsh[buf][wrow][wcb], wsrc);
  };
  stage(0, 0);
  ASYNC_WAIT();
  __syncthreads();
  int buf = 0;
  for (int kk = 0; kk < K; kk += 16) {
    if (kk + 16 < K) stage(buf ^ 1, kk + 16);  // prefetch next chunk
    compute(buf);
    ASYNC_WAIT();      // next chunk landed in LDS
    __syncthreads();   // everyone done reading current buffer
    buf ^= 1;
  }
#else
  for (int kk = 0; kk < K; kk += 16) {
    {
      const float* src = A + (size_t)(mBase + xrow) * K + kk + xcb;
      *(float4*)&xs[0][xrow][xcb] = *(const float4*)(src);
      *(float4*)&xs[0][xrow][xcb + 4] = *(const float4*)(src + 4);
    }
    {
      const float* src = W + (size_t)(nBase + wrow) * K + kk + wcb;
      *(float4*)&wsh[0][wrow][wcb] = *(const float4*)(src);
    }
    __syncthreads();
    compute(0);
    __syncthreads();
  }
#endif

  // epilogue: C layout = VGPR r -> row r (lanes 0-15) / row r+8 (lanes 16-31)
#pragma unroll
  for (int mi = 0; mi < 2; ++mi)
#pragma unroll
    for (int ni = 0; ni < 2; ++ni) {
      int col = nBase + wn * 32 + ni * 16 + l16;
      float bv = bias[col];
#pragma unroll
      for (int r = 0; r < 8; ++r) {
        int row = mBase + wm * 32 + mi * 16 + r + lh * 8;
        float val = acc[mi][ni][r] + bv;
        if (!split_heads) {
          out[(size_t)row * N + col] = val;
        } else {
          int b_ = row >> 11;  // / SEQ
          int l_ = row & (SEQ - 1);
          int h_ = col >> 6;  // / HDIM
          int d_ = col & (HDIM - 1);
          out[(((size_t)(b_ * NHEAD + h_)) * SEQ + l_) * HDIM + d_] = val;
        }
      }
    }
}

// ---------------------------------------------------------------------------
// Flash attention, fp32 WMMA. One wave owns 16 query rows (8 waves -> 128
// rows/block). Key/value processed in 32-row blocks staged through LDS
// (async global->LDS when available). Q/K/V are [B*H, L, D]; output
// merged-heads into ctx [B, L, E].
// ---------------------------------------------------------------------------
__global__ __launch_bounds__(256) void attn_flash(const float* __restrict__ Q,
                                                  const float* __restrict__ Kg,
                                                  const float* __restrict__ Vg,
                                                  float* __restrict__ ctx) {
  __shared__ float Ks[32][68];     // pad 68: conflict-free B-frag reads
  __shared__ float Vs[32][68];
  __shared__ float Ps[8][16][65];  // per-wave P transpose scratch

  const int bh = blockIdx.y;  // b*H + h
  const int b = bh >> 4, h = bh & 15;
  const int t = threadIdx.x;
  const int lane = t & 31, wid = t >> 5;
  const int l16 = lane & 15, lh = lane >> 4;
  const int qrow0 = blockIdx.x * 128 + wid * 16;
  const size_t base = (size_t)bh * SEQ * HDIM;

  // Q fragments (A layout), 1/sqrt(D) folded in
  v2f qf[16];
#pragma unroll
  for (int kc = 0; kc < 16; ++kc) {
    const float* p = Q + base + (size_t)(qrow0 + l16) * HDIM + kc * 4 + lh * 2;
    qf[kc][0] = p[0] * 0.125f;
    qf[kc][1] = p[1] * 0.125f;
  }

  v8f O[4] = {};
  float mrun[8], lrun[8];
#pragma unroll
  for (int r = 0; r < 8; ++r) {
    mrun[r] = -3.0e38f;
    lrun[r] = 0.0f;
  }

  const int krow = t >> 3, kcb = (t & 7) * 8;  // 32x64 stage, 8 floats/thr
  for (int j = 0; j < SEQ / 32; ++j) {
    const float* kp = Kg + base + (size_t)(j * 32 + krow) * HDIM + kcb;
    const float* vp = Vg + base + (size_t)(j * 32 + krow) * HDIM + kcb;
#if USE_ASYNC
    async_ld16(&Ks[krow][kcb], kp);
    async_ld16(&Ks[krow][kcb + 4], kp + 4);
    async_ld16(&Vs[krow][kcb], vp);
    async_ld16(&Vs[krow][kcb + 4], vp + 4);
    ASYNC_WAIT();
#else
    *(float4*)&Ks[krow][kcb] = *(const float4*)(kp);
    *(float4*)&Ks[krow][kcb + 4] = *(const float4*)(kp + 4);
    *(float4*)&Vs[krow][kcb] = *(const float4*)(vp);
    *(float4*)&Vs[krow][kcb + 4] = *(const float4*)(vp + 4);
#endif
    __syncthreads();

    // S = (Q*scale) @ K^T  -> two 16x16 tiles over this key block
    v8f S[2];
#pragma unroll
    for (int st = 0; st < 2; ++st) {
      v8f c = {};
#pragma unroll
      for (int kc = 0; kc < 16; ++kc) {
        int k0 = kc * 4 + lh * 2;
        v2f bf;  // B[k=d][n=s] = K[s][d]
        bf[0] = Ks[st * 16 + l16][k0];
        bf[1] = Ks[st * 16 + l16][k0 + 1];
        c = wmma4(qf[kc], bf, c);
      }
      S[st] = c;
    }

    // online softmax (row = (r, lane-half); reduce across 16-lane group)
#pragma unroll
    for (int r = 0; r < 8; ++r) {
      float mx = fmaxf(S[0][r], S[1][r]);
      mx = fmaxf(mx, __shfl_xor(mx, 1));
      mx = fmaxf(mx, __shfl_xor(mx, 2));
      mx = fmaxf(mx, __shfl_xor(mx, 4));
      mx = fmaxf(mx, __shfl_xor(mx, 8));
      float mnew = fmaxf(mrun[r], mx);
      float alpha = __expf(mrun[r] - mnew);
      S[0][r] = __expf(S[0][r] - mnew);
      S[1][r] = __expf(S[1][r] - mnew);
      float rs = S[0][r] + S[1][r];
      rs += __shfl_xor(rs, 1);
      rs += __shfl_xor(rs, 2);
      rs += __shfl_xor(rs, 4);
      rs += __shfl_xor(rs, 8);
      lrun[r] = lrun[r] * alpha + rs;
      mrun[r] = mnew;
#pragma unroll
      for (int sd = 0; sd < 4; ++sd) O[sd][r] *= alpha;
    }

    // P: C layout -> LDS -> A layout (per-wave scratch, same-wave ordering)
#pragma unroll
    for (int st = 0; st < 2; ++st)
#pragma unroll
      for (int r = 0; r < 8; ++r)
        Ps[wid][r + lh * 8][st * 16 + l16] = S[st][r];

    // O += P @ V
#pragma unroll
    for (int kc = 0; kc < 8; ++kc) {  // 32 keys / 4
      int k0 = kc * 4 + lh * 2;
      v2f a;
      a[0] = Ps[wid][l16][k0];
      a[1] = Ps[wid][l16][k0 + 1];
#pragma unroll
      for (int sd = 0; sd < 4; ++sd) {
        v2f bf;  // B[k=s][n=d] = V[s][d]
        bf[0] = Vs[k0][sd * 16 + l16];
        bf[1] = Vs[k0 + 1][sd * 16 + l16];
        O[sd] = wmma4(a, bf, O[sd]);
      }
    }
    __syncthreads();
  }

  // normalize + merge-heads store: ctx[b, l, h*64 + d]
#pragma unroll
  for (int sd = 0; sd < 4; ++sd)
#pragma unroll
    for (int r = 0; r < 8; ++r) {
      int row = qrow0 + r + lh * 8;
      int col = h * HDIM + sd * 16 + l16;
      ctx[((size_t)(b * SEQ) + row) * EMBED + col] = O[sd][r] / lrun[r];
    }
}

// ---------------------------------------------------------------------------
// LayerNorm over last dim (1024), one block per row, population variance.
// ---------------------------------------------------------------------------
__global__ __launch_bounds__(256) void layernorm_k(
    const float* __restrict__ X, const float* __restrict__ gamma,
    const float* __restrict__ beta, float* __restrict__ Y) {
  __shared__ float red[2][8];
  const int row = blockIdx.x;
  const int c = threadIdx.x * 4;
  const float* x = X + (size_t)row * EMBED;
  float4 v = *(const float4*)(x + c);
  float s = v.x + v.y + v.z + v.w;
  float sq = v.x * v.x + v.y * v.y + v.z * v.z + v.w * v.w;
#pragma unroll
  for (int m = 16; m; m >>= 1) {
    s += __shfl_xor(s, m);
    sq += __shfl_xor(sq, m);
  }
  const int lane = threadIdx.x & 31, wid = threadIdx.x >> 5;
  if (lane == 0) {
    red[0][wid] = s;
    red[1][wid] = sq;
  }
  __syncthreads();
  if (wid == 0) {
    float a = (lane < 8) ? red[0][lane] : 0.0f;
    float bsum = (lane < 8) ? red[1][lane] : 0.0f;
#pragma unroll
    for (int m = 4; m; m >>= 1) {
      a += __shfl_xor(a, m);
      bsum += __shfl_xor(bsum, m);
    }
    if (lane == 0) {
      red[0][0] = a;
      red[1][0] = bsum;
    }
  }
  __syncthreads();
  float mean = red[0][0] * (1.0f / EMBED);
  float var = red[1][0] * (1.0f / EMBED) - mean * mean;
  float rstd = rsqrtf(var + 1e-5f);
  float4 g = *(const float4*)(gamma + c);
  float4 be = *(const float4*)(beta + c);
  float4 o;
  o.x = (v.x - mean) * rstd * g.x + be.x;
  o.y = (v.y - mean) * rstd * g.y + be.y;
  o.z = (v.z - mean) * rstd * g.z + be.z;
  o.w = (v.w - mean) * rstd * g.w + be.w;
  *(float4*)(Y + (size_t)row * EMBED + c) = o;
}

// ---------------------------------------------------------------------------
extern "C" void kernel_launch(void* const* d_in, const int* in_sizes, int n_in,
                              void* d_out, int out_size, void* d_ws,
                              size_t ws_size, hipStream_t stream) {
  const float* x = (const float*)d_in[0];
  const float* wq = (const float*)d_in[1];
  const float* bq = (const float*)d_in[2];
  const float* wk = (const float*)d_in[3];
  const float* bk = (const float*)d_in[4];
  const float* wv = (const float*)d_in[5];
  const float* bv = (const float*)d_in[6];
  const float* wo = (const float*)d_in[7];
  const float* bo = (const float*)d_in[8];
  const float* ln_g = (const float*)d_in[9];
  const float* ln_b = (const float*)d_in[10];
  float* out = (float*)d_out;

  float* ws = (float*)d_ws;
  const size_t MN = (size_t)MTOT * EMBED;  // 8192*1024
  float* q = ws;
  float* k = ws + MN;
  float* v = ws + 2 * MN;
  float* ctx = ws + 3 * MN;
  float* proj = ws + 4 * MN;

  dim3 gg(EMBED / 64, MTOT / 128);  // (16, 64)
  gemm_wmma<<<gg, 256, 0, stream>>>(x, wq, bq, q, MTOT, EMBED, EMBED, 1);
  gemm_wmma<<<gg, 256, 0, stream>>>(x, wk, bk, k, MTOT, EMBED, EMBED, 1);
  gemm_wmma<<<gg, 256, 0, stream>>>(x, wv, bv, v, MTOT, EMBED, EMBED, 1);

  attn_flash<<<dim3(SEQ / 128, BATCH * NHEAD), 256, 0, stream>>>(q, k, v, ctx);

  gemm_wmma<<<gg, 256, 0, stream>>>(ctx, wo, bo, proj, MTOT, EMBED, EMBED, 0);
  layernorm_k<<<MTOT, 256, 0, stream>>>(proj, ln_g, ln_b, out);
}